// HGNN_55516747268119
// MI455X (gfx1250) — compile-verified
//
#include <hip/hip_runtime.h>

#define N_NODES 100000
#define N_EDGES 1600000
#define F 64
#define MIN_NORM 1e-15f
#define MAXNORM 0.996f   /* (1 - 4e-3)/sqrt(c), c = 1 */

typedef float v2f __attribute__((ext_vector_type(2)));
typedef float v8f __attribute__((ext_vector_type(8)));

__device__ __forceinline__ float artanhf_(float x) {
    x = fminf(fmaxf(x, -1.0f + 1e-7f), 1.0f - 1e-7f);
    return 0.5f * (log1pf(x) - log1pf(-x));
}

// ---------------------------------------------------------------------------
// encode: out = proj(expmap0(x))   (2 lanes per node, 32 feats per lane)
// ---------------------------------------------------------------------------
__global__ __launch_bounds__(256) void k_map0(const float* __restrict__ in,
                                              float* __restrict__ out) {
    int gid  = blockIdx.x * 256 + threadIdx.x;
    int node = gid >> 1;
    if (node >= N_NODES) return;
    int half = gid & 1;
    const float* p = in + node * F + half * 32;
    float v[32];
    float s = 0.f;
#pragma unroll
    for (int i = 0; i < 32; ++i) { v[i] = p[i]; s += v[i] * v[i]; }
    s += __shfl_xor(s, 1, 32);
    float n  = sqrtf(fmaxf(s, MIN_NORM));
    float ef = tanhf(n) / n;                               // expmap0 scale
    float en = sqrtf(fmaxf(ef * ef * s, MIN_NORM));
    float sc = ef * (en > MAXNORM ? MAXNORM / en : 1.0f);  // + proj clip
    float* q = out + node * F + half * 32;
#pragma unroll
    for (int i = 0; i < 32; ++i) q[i] = sc * v[i];
}

// ---------------------------------------------------------------------------
// hyp_linear + logmap0 fused:  xt = logmap0(proj(mobius_add(proj(mobius_matvec(
//     W, h)), hyp_b)))    One wave per 16-node tile; GEMM via WMMA f32 16x16x4.
// ---------------------------------------------------------------------------
__global__ __launch_bounds__(256) void k_hyp_linear(const float* __restrict__ h,
                                                    const float* __restrict__ Wm,
                                                    const float* __restrict__ bias,
                                                    float* __restrict__ xt) {
    __shared__ float smx[8][16][64];   // per-wave mx staging (32 KB)

    const int lane = threadIdx.x & 31;
    const int wv   = threadIdx.x >> 5;
    const int l15  = lane & 15;
    const int koff = (lane >> 4) << 1;     // lanes 16-31 hold K+2,K+3

    int base = (blockIdx.x * 8 + wv) * 16;
    if (base > N_NODES - 16) base = N_NODES - 16;   // clamp: duplicate identical work
    const int node = base + l15;

    // ---- load A fragments (16x64 node tile) + per-node input norm ----
    v2f   av[16];
    float xs = 0.f;
#pragma unroll
    for (int s = 0; s < 16; ++s) {
        av[s] = *(const v2f*)(h + node * F + 4 * s + koff);
        xs += av[s][0] * av[s][0] + av[s][1] * av[s][1];
    }
    float xn2v = xs + __shfl_xor(xs, 16, 32);  // lane l: ||h[base+(l&15)]||^2

    // ---- mx = h @ W^T via 64 x v_wmma_f32_16x16x4_f32 ----
#pragma unroll
    for (int nt = 0; nt < 4; ++nt) {
        v8f c = {};
        const float* wrow = Wm + (nt * 16 + l15) * F + koff;  // B[k][o] = W[o][k]
#pragma unroll
        for (int s = 0; s < 16; ++s) {
            v2f b = *(const v2f*)(wrow + 4 * s);
            c = __builtin_amdgcn_wmma_f32_16x16x4_f32(false, av[s], false, b,
                                                      (short)0, c, false, false);
        }
#pragma unroll
        for (int j = 0; j < 8; ++j) {   // C layout: VGPR j -> M=j (+8 for hi lanes)
            int m = j + ((lane >> 4) << 3);
            smx[wv][m][nt * 16 + l15] = c[j];
        }
    }
    __syncthreads();

    // ---- hyperbolic bias hyp_b = proj(expmap0(b)) (scalar, same all nodes) ----
    float bn2 = 0.f;
    for (int f = 0; f < F; ++f) { float bv = bias[f]; bn2 += bv * bv; }
    float bn  = sqrtf(fmaxf(bn2, MIN_NORM));
    float bf  = tanhf(bn) / bn;
    float hbn = sqrtf(fmaxf(bf * bf * bn2, MIN_NORM));
    float bs  = bf * (hbn > MAXNORM ? MAXNORM / hbn : 1.0f);
    float y2  = bs * bs * bn2;

    // ---- per-node tail: lanes (2p, 2p+1) handle node p, 32 feats each ----
    const int p    = lane >> 1;
    const int half = lane & 1;
    const int f0   = half * 32;

    float xn2 = __shfl(xn2v, p, 32);
    float xn  = sqrtf(fmaxf(xn2, MIN_NORM));

    float ms = 0.f;
    for (int i = 0; i < 32; ++i) { float m = smx[wv][p][f0 + i]; ms += m * m; }
    float mxn2 = ms + __shfl_xor(ms, 1, 32);

    // mobius_matvec scale + proj
    float mxn  = sqrtf(fmaxf(mxn2, MIN_NORM));
    float fac1 = tanhf(mxn / xn * artanhf_(xn)) / mxn;
    if (mxn2 == 0.f) fac1 = 0.f;                       // all-zero mx guard
    float rn = sqrtf(fmaxf(fac1 * fac1 * mxn2, MIN_NORM));
    float s1 = fac1 * (rn > MAXNORM ? MAXNORM / rn : 1.0f);
    float x2 = s1 * s1 * mxn2;

    // <res, hyp_b>
    float dsum = 0.f;
    for (int i = 0; i < 32; ++i) dsum += smx[wv][p][f0 + i] * bias[f0 + i];
    dsum += __shfl_xor(dsum, 1, 32);
    float xy = s1 * bs * dsum;

    // mobius_add (c=1), proj, logmap0 — all analytic
    float al  = 1.f + 2.f * xy + y2;
    float be  = 1.f - x2;
    float den = fmaxf(1.f + 2.f * xy + x2 * y2, MIN_NORM);
    float n2  = (al * al * x2 + 2.f * al * be * xy + be * be * y2) / (den * den);
    float nn  = sqrtf(fmaxf(n2, MIN_NORM));
    float s2  = (nn > MAXNORM ? MAXNORM / nn : 1.0f);
    float nf  = fminf(nn, MAXNORM);
    float lf  = artanhf_(nf) / nf;

    float c1 = lf * s2 * al * s1 / den;   // coefficient on mx
    float c2 = lf * s2 * be * bs / den;   // coefficient on raw bias
    float* q = xt + (size_t)(base + p) * F;
    for (int i = 0; i < 32; ++i)
        q[f0 + i] = c1 * smx[wv][p][f0 + i] + c2 * bias[f0 + i];
}

// ---------------------------------------------------------------------------
// zero scratch
// ---------------------------------------------------------------------------
__global__ __launch_bounds__(256) void k_zero(float* __restrict__ p, int n) {
    int i = blockIdx.x * 256 + threadIdx.x;
    if (i < n) p[i] = 0.f;
}

// ---------------------------------------------------------------------------
// edge aggregation: agg[dst] += w * xt[src]   (32 lanes/edge, 2 feats/lane)
// ---------------------------------------------------------------------------
__global__ __launch_bounds__(256) void k_agg(const float* __restrict__ xt,
                                             const int* __restrict__ src,
                                             const int* __restrict__ dst,
                                             const float* __restrict__ ew,
                                             float* __restrict__ agg) {
    int e = blockIdx.x * 8 + (threadIdx.x >> 5);
    if (e >= N_EDGES) return;
    int   f = threadIdx.x & 31;
    int   s = src[e], d = dst[e];
    float w = ew[e];
    atomicAdd(&agg[(size_t)d * F + f],      xt[(size_t)s * F + f]      * w);
    atomicAdd(&agg[(size_t)d * F + f + 32], xt[(size_t)s * F + f + 32] * w);
}

// ---------------------------------------------------------------------------
// post-agg: h' = proj(expmap0(relu(logmap0(proj(expmap0(agg))))))
// ---------------------------------------------------------------------------
__global__ __launch_bounds__(256) void k_post(const float* __restrict__ agg,
                                              float* __restrict__ out) {
    int gid  = blockIdx.x * 256 + threadIdx.x;
    int node = gid >> 1;
    if (node >= N_NODES) return;
    int half = gid & 1;
    const float* p = agg + node * F + half * 32;
    float v[32];
    float s = 0.f, sp = 0.f;
#pragma unroll
    for (int i = 0; i < 32; ++i) {
        v[i] = p[i];
        float t = v[i] * v[i];
        s += t;
        if (v[i] > 0.f) sp += t;   // positive components survive relu
    }
    s  += __shfl_xor(s, 1, 32);
    sp += __shfl_xor(sp, 1, 32);

    float un = sqrtf(fmaxf(s, MIN_NORM));
    float ef = tanhf(un) / un;                              // expmap0
    float en = sqrtf(fmaxf(ef * ef * s, MIN_NORM));
    float es = ef * (en > MAXNORM ? MAXNORM / en : 1.0f);   // proj
    float hn = sqrtf(fmaxf(es * es * s, MIN_NORM));
    float lfv = artanhf_(hn) / hn;                          // logmap0
    float scale = es * lfv;                                 // tangent scale (>0)
    float rn2 = scale * scale * sp;                         // ||relu(t)||^2
    float rnn = sqrtf(fmaxf(rn2, MIN_NORM));
    float ff  = tanhf(rnn) / rnn;                           // expmap0
    float fn  = sqrtf(fmaxf(ff * ff * rn2, MIN_NORM));
    float fs  = ff * (fn > MAXNORM ? MAXNORM / fn : 1.0f);  // proj

    float* q = out + node * F + half * 32;
#pragma unroll
    for (int i = 0; i < 32; ++i) q[i] = fs * fmaxf(0.f, scale * v[i]);
}

// ---------------------------------------------------------------------------
extern "C" void kernel_launch(void* const* d_in, const int* in_sizes, int n_in,
                              void* d_out, int out_size, void* d_ws, size_t ws_size,
                              hipStream_t stream) {
    const float* x  = (const float*)d_in[0];
    const int*   ei = (const int*)d_in[1];
    const float* ew = (const float*)d_in[2];
    const float* W0 = (const float*)d_in[3];
    const float* b0 = (const float*)d_in[4];
    const float* W1 = (const float*)d_in[5];
    const float* b1 = (const float*)d_in[6];
    const int* src = ei;
    const int* dst = ei + N_EDGES;

    float* bufA = (float*)d_ws;
    float* bufB = bufA + (size_t)N_NODES * F;
    float* out  = (float*)d_out;

    dim3 blk(256);
    int gElem = (2 * N_NODES + 255) / 256;
    int gTile = (N_NODES / 16 + 7) / 8;
    int gZero = (N_NODES * F + 255) / 256;
    int gEdge = (N_EDGES + 7) / 8;

    k_map0<<<gElem, blk, 0, stream>>>(x, bufA);

    // layer 0
    k_hyp_linear<<<gTile, blk, 0, stream>>>(bufA, W0, b0, bufB);  // -> xt
    k_zero<<<gZero, blk, 0, stream>>>(bufA, N_NODES * F);
    k_agg<<<gEdge, blk, 0, stream>>>(bufB, src, dst, ew, bufA);   // -> agg
    k_post<<<gElem, blk, 0, stream>>>(bufA, bufB);                // -> h1

    // layer 1
    k_hyp_linear<<<gTile, blk, 0, stream>>>(bufB, W1, b1, bufA);
    k_zero<<<gZero, blk, 0, stream>>>(bufB, N_NODES * F);
    k_agg<<<gEdge, blk, 0, stream>>>(bufA, src, dst, ew, bufB);
    k_post<<<gElem, blk, 0, stream>>>(bufB, out);
}